// VectorQuantizer_19310172963143
// MI455X (gfx1250) — compile-verified
//
#include <hip/hip_runtime.h>
#include <hip/hip_bf16.h>
#include <math.h>

typedef __attribute__((ext_vector_type(2))) float v2f;
typedef __attribute__((ext_vector_type(8))) float v8f;

#define NUM_EMB 512
#define EMB_DIM 64
#define NROWS   131072      /* 32*64*64 flattened NHWC rows */
#define HW      4096        /* 64*64 */
#define ROWS_PB 16

/* d_out flat layout (return-order concat, all float32):
   loss(1) | quantized NCHW(8388608) | perplexity(1) | distances(67108864)
   | encoding_indices(131072) | encodings(67108864)                      */
#define OFF_LOSS   0
#define OFF_QUANT  1
#define OFF_PERP   8388609
#define OFF_DIST   8388610
#define OFF_IDX    75497474
#define OFF_ENC    75628546

__global__ __launch_bounds__(512) void vq_init(float* __restrict__ out,
                                               unsigned int* __restrict__ hist) {
    const int t = threadIdx.x;
    if (t == 0) out[OFF_LOSS] = 0.0f;
    if (t == 1) out[OFF_PERP] = 0.0f;
    if (t < NUM_EMB) hist[t] = 0u;
}

__global__ __launch_bounds__(256) void vq_main(const float* __restrict__ in,
                                               const float* __restrict__ emb,
                                               float* __restrict__ out,
                                               unsigned int* __restrict__ hist) {
    __shared__ float lx[ROWS_PB * 65];   /* x tile, stride 65 (bank-conflict pad) */
    __shared__ float lesq[NUM_EMB];      /* ||e||^2 per code                      */
    __shared__ float lxsq[ROWS_PB];      /* ||x||^2 per row                       */
    __shared__ float lmv[2048];          /* argmin candidates: [row16][wave8*ln16]*/
    __shared__ int   lmi[2048];
    __shared__ int   lwidx[ROWS_PB];     /* winning code per row                  */
    __shared__ float lred[256];

    const int tid = threadIdx.x;
    const int wv  = tid >> 5;        /* wave 0..7   */
    const int ln  = tid & 31;        /* lane 0..31  */
    const int m   = ln & 15;
    const int hf  = ln >> 4;

    const int rowBase = blockIdx.x * ROWS_PB;
    const int n  = rowBase >> 12;          /* batch index          */
    const int hw = rowBase & (HW - 1);     /* h*64+w base (mult16) */

    /* stage the 16x64 x-tile from NCHW into LDS (coalesced over r) */
    #pragma unroll
    for (int j = 0; j < 4; ++j) {
        const int item = tid + j * 256;      /* 0..1023 */
        const int c = item >> 4, r = item & 15;
        lx[r * 65 + c] = in[(n * EMB_DIM + c) * HW + hw + r];
    }
    /* codebook squared norms (codebook is L2-resident) */
    #pragma unroll
    for (int j = 0; j < 2; ++j) {
        const int code = tid + j * 256;
        const float* e = emb + code * EMB_DIM;
        float s = 0.0f;
        #pragma unroll
        for (int c = 0; c < EMB_DIM; ++c) s += e[c] * e[c];
        lesq[code] = s;
    }
    __syncthreads();
    if (tid < ROWS_PB) {
        float s = 0.0f;
        #pragma unroll
        for (int c = 0; c < EMB_DIM; ++c) { const float v = lx[tid * 65 + c]; s += v * v; }
        lxsq[tid] = s;
    }
    __syncthreads();

    /* A fragments: 32-bit A 16x4 layout — lane m = row M, VGPR pair holds
       K = 4s + 2*hf, 4s + 2*hf + 1 */
    v2f a[16];
    #pragma unroll
    for (int s = 0; s < 16; ++s) {
        const int k = 4 * s + 2 * hf;
        v2f t; t.x = lx[m * 65 + k]; t.y = lx[m * 65 + k + 1];
        a[s] = t;
    }

    float bestv[8];
    int   besti[8];
    #pragma unroll
    for (int v = 0; v < 8; ++v) { bestv[v] = 3.4e38f; besti[v] = 0; }

    float* __restrict__ dist = out + OFF_DIST;

    /* each wave covers 4 code tiles of 16 -> 8 waves * 64 codes = 512 */
    for (int j = 0; j < 4; ++j) {
        const int codeBase = (wv * 4 + j) * 16;
        /* B fragments: B[k][nc] = emb[nc][k]; same striping as A with code row */
        const float* ebase = emb + (codeBase + m) * EMB_DIM + 2 * hf;
        v2f b[16];
        #pragma unroll
        for (int s = 0; s < 16; ++s)
            b[s] = *(const v2f*)(ebase + 4 * s);     /* 8B aligned */

        v8f acc = {};
        #pragma unroll
        for (int s = 0; s < 16; ++s)
            acc = __builtin_amdgcn_wmma_f32_16x16x4_f32(
                false, a[s], false, b[s], (short)0, acc, false, false);

        const int   code = codeBase + m;
        const float eq   = lesq[code];
        #pragma unroll
        for (int v = 0; v < 8; ++v) {
            const int rloc = hf * 8 + v;               /* C layout: M = v + 8*hf */
            const float dval = lxsq[rloc] + eq - 2.0f * acc[v];
            dist[(size_t)(rowBase + rloc) * NUM_EMB + code] = dval;
            if (dval < bestv[v]) { bestv[v] = dval; besti[v] = code; }
        }
    }

    /* per-lane candidates -> LDS */
    #pragma unroll
    for (int v = 0; v < 8; ++v) {
        const int rloc = hf * 8 + v;
        lmv[rloc * 128 + wv * 16 + m] = bestv[v];
        lmi[rloc * 128 + wv * 16 + m] = besti[v];
    }
    __syncthreads();

    if (tid < ROWS_PB) {
        float bv = 3.4e38f; int bi = 0x7fffffff;
        for (int q = 0; q < 128; ++q) {
            const float v = lmv[tid * 128 + q];
            const int   i = lmi[tid * 128 + q];
            if (v < bv || (v == bv && i < bi)) { bv = v; bi = i; }
        }
        lwidx[tid] = bi;
        out[OFF_IDX + rowBase + tid] = (float)bi;
        atomicAdd(&hist[bi], 1u);
    }
    __syncthreads();

    /* quantized (NCHW scatter) + fused loss partial:
       loss = (1 + 0.25) * mean((q - x)^2)  (e/q latent losses equal in value) */
    float lsum = 0.0f;
    #pragma unroll
    for (int j = 0; j < 4; ++j) {
        const int item = tid + j * 256;
        const int c = item >> 4, r = item & 15;
        const float q = emb[lwidx[r] * EMB_DIM + c];
        const float x = lx[r * 65 + c];
        const float d = q - x;
        lsum += d * d;
        out[OFF_QUANT + (n * EMB_DIM + c) * HW + hw + r] = q;
    }

    /* one-hot encodings: 16 rows * 512 = 8192 floats, fully coalesced */
    float* __restrict__ enc = out + OFF_ENC + (size_t)rowBase * NUM_EMB;
    #pragma unroll
    for (int j = 0; j < 32; ++j) {
        const int item = tid + j * 256;              /* == r*512 + code */
        const int r = item >> 9, code = item & (NUM_EMB - 1);
        enc[item] = (code == lwidx[r]) ? 1.0f : 0.0f;
    }

    /* block loss reduction */
    lred[tid] = lsum;
    __syncthreads();
    for (int s = 128; s > 0; s >>= 1) {
        if (tid < s) lred[tid] += lred[tid + s];
        __syncthreads();
    }
    if (tid == 0)
        atomicAdd(out + OFF_LOSS, lred[0] * (1.25f / 8388608.0f));
}

__global__ __launch_bounds__(512) void vq_perp(const unsigned int* __restrict__ hist,
                                               float* __restrict__ out) {
    __shared__ float red[NUM_EMB];
    const int t = threadIdx.x;
    const float p = (float)hist[t] * (1.0f / (float)NROWS);
    red[t] = p * logf(p + 1e-10f);
    __syncthreads();
    for (int s = 256; s > 0; s >>= 1) {
        if (t < s) red[t] += red[t + s];
        __syncthreads();
    }
    if (t == 0) out[OFF_PERP] = expf(-red[0]);
}

extern "C" void kernel_launch(void* const* d_in, const int* in_sizes, int n_in,
                              void* d_out, int out_size, void* d_ws, size_t ws_size,
                              hipStream_t stream) {
    const float* in  = (const float*)d_in[0];   /* (32,64,64,64) NCHW fp32 */
    const float* emb = (const float*)d_in[1];   /* (512,64) fp32           */
    float* out = (float*)d_out;
    unsigned int* hist = (unsigned int*)d_ws;   /* 512 counters            */

    vq_init<<<1, 512, 0, stream>>>(out, hist);
    vq_main<<<NROWS / ROWS_PB, 256, 0, stream>>>(in, emb, out, hist);
    vq_perp<<<1, 512, 0, stream>>>(hist, out);
}